// AutoEncoder_4647154614489
// MI455X (gfx1250) — compile-verified
//
#include <hip/hip_runtime.h>
#include <stdint.h>

#define FEAT 192
#define ROW_BYTES (FEAT * 4)                       // 768 B per corner row
#define CORNERS_PER_TILE 8
#define WAVES_PER_BLOCK 8
#define THREADS 256
#define TILE_BYTES (CORNERS_PER_TILE * ROW_BYTES)  // 6144 B per wave tile
#define CHUNKS_PER_LANE (TILE_BYTES / 16 / 32)     // 12 b128 issues per lane
#define FLOATS_PER_LANE (FEAT / 32)                // 6 floats per lane per corner
#define TILE_FLOATS (CORNERS_PER_TILE * FEAT)      // 1536

// ---------------------------------------------------------------------------
// Kernel 1: zero sums (d_out) and counts (d_ws), float4-vectorized.
// out_size (V*192) and V are both divisible by 4.
// ---------------------------------------------------------------------------
__global__ void zero_kernel(float4* __restrict__ out4, float4* __restrict__ cnt4,
                            int nOut4, int nCnt4) {
    const int total = nOut4 + nCnt4;
    const float4 z = make_float4(0.f, 0.f, 0.f, 0.f);
    for (int i = blockIdx.x * blockDim.x + threadIdx.x; i < total;
         i += gridDim.x * blockDim.x) {
        if (i < nOut4) out4[i] = z;
        else           cnt4[i - nOut4] = z;
    }
}

// ---------------------------------------------------------------------------
// Kernel 2: double-buffered async global->LDS staging (ASYNCcnt pipeline),
// then scatter-add with global f32 atomics (resolve in the 192MB L2).
// ---------------------------------------------------------------------------
__global__ __launch_bounds__(THREADS)
void scatter_add_kernel(const float* __restrict__ feats,
                        const int* __restrict__ faces,
                        float* __restrict__ sums,
                        float* __restrict__ counts,
                        int nCorners) {
    // [wave][buffer][tile floats] : 8 * 2 * 6KB = 96KB per block
    __shared__ float smem[WAVES_PER_BLOCK][2][TILE_FLOATS];

    const int wave  = threadIdx.x >> 5;   // wave32
    const int lane  = threadIdx.x & 31;
    const int gwave = blockIdx.x * WAVES_PER_BLOCK + wave;
    const int nWaves = gridDim.x * WAVES_PER_BLOCK;
    const int nTiles = (nCorners + CORNERS_PER_TILE - 1) / CORNERS_PER_TILE;

    // Low 32 bits of a generic LDS pointer == LDS byte offset (AS3 offset).
    const uint32_t ldsBuf0 = (uint32_t)(uintptr_t)(&smem[wave][0][0]);
    const uint32_t ldsBuf1 = (uint32_t)(uintptr_t)(&smem[wave][1][0]);
    const char* featsBytes = (const char*)feats;
    const long long lastChunk = (long long)nCorners * ROW_BYTES - 16;

    // ---- stage one tile (always exactly 12 b128 issues, full EXEC) ----
    auto stage = [&](int tile, uint32_t ldsBase) {
        const long long tileByte = (long long)tile * TILE_BYTES;
        #pragma unroll
        for (int k = 0; k < CHUNKS_PER_LANE; ++k) {
            const int chunk = k * 32 + lane;
            long long gb = tileByte + (long long)chunk * 16;
            if (gb > lastChunk) gb = lastChunk;     // clamp: tail garbage never read
            const void* gptr = (const void*)(featsBytes + gb);
            const uint32_t laddr = ldsBase + (uint32_t)(chunk * 16);
            asm volatile("global_load_async_to_lds_b128 %0, %1, off"
                         :: "v"(laddr), "v"(gptr) : "memory");
        }
    };

    auto process = [&](int tile, const float* buf) {
        const int cornerBase = tile * CORNERS_PER_TILE;
        #pragma unroll
        for (int c = 0; c < CORNERS_PER_TILE; ++c) {
            const int corner = cornerBase + c;
            if (corner >= nCorners) break;          // wave-uniform guard
            const int v = faces[corner];            // broadcast load
            const float* src = buf + c * FEAT + lane * FLOATS_PER_LANE;
            float* dst = sums + (long long)v * FEAT + lane * FLOATS_PER_LANE;
            #pragma unroll
            for (int j = 0; j < FLOATS_PER_LANE; ++j)
                atomicAdd(dst + j, src[j]);         // global_atomic_add_f32
            if (lane == 0)
                atomicAdd(counts + v, 1.0f);
        }
    };

    int tile = gwave;
    if (tile >= nTiles) return;

    // prologue: stage first tile into buffer 0
    stage(tile, ldsBuf0);
    int buf = 0;

    while (tile < nTiles) {
        const int next = tile + nWaves;
        const bool hasNext = next < nTiles;
        if (hasNext) {
            stage(next, buf ? ldsBuf0 : ldsBuf1);
            // 12 copies outstanding for `next`; current tile's 12 are older and
            // complete in order -> waiting for <=12 drains the current tile.
            asm volatile("s_wait_asynccnt 0xc" ::: "memory");
        } else {
            asm volatile("s_wait_asynccnt 0x0" ::: "memory");
        }
        process(tile, &smem[wave][buf][0]);
        buf ^= 1;
        tile = next;
    }
}

// ---------------------------------------------------------------------------
// Kernel 3: divide each vertex row by its count (float4 vectorized)
// ---------------------------------------------------------------------------
__global__ void divide_kernel(float4* __restrict__ out4,
                              const float* __restrict__ counts, int n4) {
    for (int i = blockIdx.x * blockDim.x + threadIdx.x; i < n4;
         i += gridDim.x * blockDim.x) {
        const float c = counts[i / (FEAT / 4)];     // 48 float4 per row
        float4 s = out4[i];
        s.x /= c; s.y /= c; s.z /= c; s.w /= c;
        out4[i] = s;
    }
}

// ---------------------------------------------------------------------------
extern "C" void kernel_launch(void* const* d_in, const int* in_sizes, int n_in,
                              void* d_out, int out_size, void* d_ws, size_t ws_size,
                              hipStream_t stream) {
    const float* feats = (const float*)d_in[0];   // [F, 576] f32
    const int*   faces = (const int*)d_in[1];     // [F, 3] int
    const int nCorners = in_sizes[1];             // 3F
    const int V        = out_size / FEAT;         // vertex_count
    float* out    = (float*)d_out;                // [V, 192] f32 (sums -> mean)
    float* counts = (float*)d_ws;                 // [V] f32 scratch

    // 1) zero accumulators (float4 stores)
    {
        int nOut4 = out_size / 4, nCnt4 = V / 4;
        int total = nOut4 + nCnt4;
        int blocks = (total + THREADS - 1) / THREADS;
        if (blocks > 4096) blocks = 4096;
        zero_kernel<<<blocks, THREADS, 0, stream>>>((float4*)out, (float4*)counts,
                                                    nOut4, nCnt4);
    }

    // 2) scatter-add (double-buffered async LDS staging + f32 atomics)
    {
        int nTiles = (nCorners + CORNERS_PER_TILE - 1) / CORNERS_PER_TILE;
        int blocks = (nTiles + WAVES_PER_BLOCK - 1) / WAVES_PER_BLOCK;
        if (blocks > 3072) blocks = 3072;   // grid-stride: waves run several tiles
        scatter_add_kernel<<<blocks, THREADS, 0, stream>>>(feats, faces, out,
                                                           counts, nCorners);
    }

    // 3) divide by counts
    {
        int n4 = out_size / 4;
        int blocks = (n4 + THREADS - 1) / THREADS;
        if (blocks > 4096) blocks = 4096;
        divide_kernel<<<blocks, THREADS, 0, stream>>>((float4*)out, counts, n4);
    }
}